// Pixelwise_15066745274518
// MI455X (gfx1250) — compile-verified
//
#include <hip/hip_runtime.h>
#include <math.h>

typedef float v2f __attribute__((ext_vector_type(2)));
typedef float v8f __attribute__((ext_vector_type(8)));

#define NN   10000
#define KK   3
#define NPIX 32768
#define CHUNK 2000   // depths per LDS chunk (32 KB), 5 chunks, 125 wmma iters each

#if defined(__has_builtin)
#  if __has_builtin(__builtin_amdgcn_global_load_async_to_lds_b128)
#    define HAVE_ASYNC_LDS 1
#  endif
#endif
#ifndef HAVE_ASYNC_LDS
#  define HAVE_ASYNC_LDS 0
#endif

#if HAVE_ASYNC_LDS
// exact parameter types per hipcc diagnostic: V4i with AS1 (global) / AS3 (LDS)
typedef int v4i_vs __attribute__((vector_size(16)));
typedef __attribute__((address_space(1))) v4i_vs* as1_v4i;
typedef __attribute__((address_space(3))) v4i_vs* as3_v4i;
#endif

// ---- constants from the reference ----
constexpr double kC_LIGHT = 299792458.0 * 1000.0;     // mm/s
constexpr double kF_MAX   = kC_LIGHT / (2.0 * 10000.0);
constexpr double kTAU     = 1.0 / kF_MAX;             // tau = tauMin
constexpr double kDT      = kTAU / (double)NN;
constexpr double kP_SRC   = 1.0e9;
constexpr double kP_AMB   = 1.0e6;
constexpr double kBETA    = 1.0e-4;
constexpr double kT_INT   = 0.1;
constexpr double kGAMMA   = 1.0 / (kBETA * kT_INT * (kP_AMB + kP_SRC));
constexpr double kC1      = kGAMMA * kBETA * (kT_INT / kTAU);   // b_mean scale
constexpr double kRNG2    = (20.0 * kGAMMA) * (20.0 * kGAMMA);  // (READ_NOISE*GAMMA)^2

// ---------------------------------------------------------------------------
// Kernel A: sum(ModFs), kappas[k] = sum(DemodFs[:,k]) * DT   -> stats[0..3]
// ---------------------------------------------------------------------------
__global__ __launch_bounds__(256) void stats_kernel(const float* __restrict__ mod,
                                                    const float* __restrict__ demod,
                                                    float* __restrict__ stats) {
    __shared__ float red[4][256];
    int tid = threadIdx.x;
    float sm = 0.f, s0 = 0.f, s1 = 0.f, s2 = 0.f;
    for (int i = tid; i < NN; i += 256) {
        sm += mod[i];
        s0 += demod[i * 3 + 0];
        s1 += demod[i * 3 + 1];
        s2 += demod[i * 3 + 2];
    }
    red[0][tid] = sm; red[1][tid] = s0; red[2][tid] = s1; red[3][tid] = s2;
    __syncthreads();
    for (int s = 128; s > 0; s >>= 1) {
        if (tid < s) {
            red[0][tid] += red[0][tid + s];
            red[1][tid] += red[1][tid + s];
            red[2][tid] += red[2][tid + s];
            red[3][tid] += red[3][tid + s];
        }
        __syncthreads();
    }
    if (tid == 0) {
        stats[0] = red[0][0];
        stats[1] = red[1][0] * (float)kDT;
        stats[2] = red[2][0] * (float)kDT;
        stats[3] = red[3][0] * (float)kDT;
    }
}

// ---------------------------------------------------------------------------
// Kernel B: corr[n,k] = DT*scale * sum_m Mod[m]*Demod[(m+n)%N,k]; then
// per-row normalize over K (ddof=1).  Writes corrW[n*4+k] and normW[n*4+k].
// ---------------------------------------------------------------------------
__global__ __launch_bounds__(256) void corr_kernel(const float* __restrict__ mod,
                                                   const float* __restrict__ demod,
                                                   const float* __restrict__ stats,
                                                   float* __restrict__ corrW,
                                                   float* __restrict__ normW) {
    __shared__ float smod[NN];                     // 40 KB
    for (int i = threadIdx.x; i < NN; i += blockDim.x) smod[i] = mod[i];
    __syncthreads();

    int n = blockIdx.x * blockDim.x + threadIdx.x;
    if (n >= NN) return;

    float a0 = 0.f, a1 = 0.f, a2 = 0.f;
    int idx = n;
    for (int m = 0; m < NN; ++m) {
        float wm = smod[m];
        const float* dp = demod + idx * 3;
        a0 = fmaf(wm, dp[0], a0);
        a1 = fmaf(wm, dp[1], a1);
        a2 = fmaf(wm, dp[2], a2);
        idx++; if (idx >= NN) idx = 0;
    }
    float f  = ((float)kP_SRC * (float)NN / stats[0]) * (float)kDT;
    float c0 = a0 * f, c1 = a1 * f, c2 = a2 * f;
    float mu = (c0 + c1 + c2) * (1.0f / 3.0f);
    float d0 = c0 - mu, d1 = c1 - mu, d2 = c2 - mu;
    float inv = rsqrtf((d0 * d0 + d1 * d1 + d2 * d2) * 0.5f);   // ddof=1
    corrW[n * 4 + 0] = c0; corrW[n * 4 + 1] = c1;
    corrW[n * 4 + 2] = c2; corrW[n * 4 + 3] = 0.f;
    normW[n * 4 + 0] = d0 * inv; normW[n * 4 + 1] = d1 * inv;
    normW[n * 4 + 2] = d2 * inv; normW[n * 4 + 3] = 0.f;
}

// ---------------------------------------------------------------------------
// Kernel C: per-pixel brightness model + clip + normalize -> normB[p*4+k]
// ---------------------------------------------------------------------------
__global__ __launch_bounds__(256) void pix_kernel(const int* __restrict__ gt,
                                                  const float* __restrict__ noise,
                                                  const float* __restrict__ stats,
                                                  const float* __restrict__ corrW,
                                                  float* __restrict__ normB) {
    int p = blockIdx.x * blockDim.x + threadIdx.x;
    if (p >= NPIX) return;
    int g = gt[p];
    float b[3];
#pragma unroll
    for (int k = 0; k < 3; ++k) {
        float c   = corrW[g * 4 + k];
        float bm  = (float)kC1 * (c + (float)kP_AMB * stats[1 + k]);
        float var = bm * (float)kGAMMA + (float)kRNG2;
        float bb  = fmaf(sqrtf(var), noise[p * 3 + k], bm);
        b[k] = fminf(fmaxf(bb, 0.f), 1.f);
    }
    float mu = (b[0] + b[1] + b[2]) * (1.0f / 3.0f);
    float d0 = b[0] - mu, d1 = b[1] - mu, d2 = b[2] - mu;
    float inv = rsqrtf((d0 * d0 + d1 * d1 + d2 * d2) * 0.5f);   // ddof=1
    normB[p * 4 + 0] = d0 * inv; normB[p * 4 + 1] = d1 * inv;
    normB[p * 4 + 2] = d2 * inv; normB[p * 4 + 3] = 0.f;
}

// ---------------------------------------------------------------------------
// Helpers for the WMMA argmax kernel
// ---------------------------------------------------------------------------
__device__ __forceinline__ void wait_async0() {
#if HAVE_ASYNC_LDS
#  if __has_builtin(__builtin_amdgcn_s_wait_asynccnt)
    __builtin_amdgcn_s_wait_asynccnt(0);
#  else
    asm volatile("s_wait_asynccnt 0" ::: "memory");
#  endif
#endif
}

__device__ __forceinline__ void stage_chunk(const float4* __restrict__ src,
                                            float4* __restrict__ dst, int tid) {
#if HAVE_ASYNC_LDS
    for (int i = tid; i < CHUNK; i += 256) {
        __builtin_amdgcn_global_load_async_to_lds_b128(
            (as1_v4i)(src + i), (as3_v4i)(dst + i),
            /*offset=*/0, /*cpol=*/0);
    }
#else
    for (int i = tid; i < CHUNK; i += 256) dst[i] = src[i];
#endif
}

// 125 WMMAs over one staged chunk, 5-way batched so the 5 ds_load_b64 issue
// ahead of the 5 v_wmma (hides DScnt wait).
__device__ __forceinline__ void wmma_chunk(const float* __restrict__ lbase,
                                           v2f A, int ln, int koff, int depthBase,
                                           float best[8], int bidx[8]) {
    for (int j = 0; j < CHUNK / 16; j += 5) {
        v2f Bv[5];
#pragma unroll
        for (int u = 0; u < 5; ++u) {
            const float* bp = lbase + ((j + u) * 16 + ln) * 4 + koff;
            Bv[u].x = bp[0];
            Bv[u].y = bp[1];
        }
#pragma unroll
        for (int u = 0; u < 5; ++u) {
            v8f Cc = {};
            v8f D = __builtin_amdgcn_wmma_f32_16x16x4_f32(
                false, A, false, Bv[u], (short)0, Cc, false, false);
            int depth = depthBase + (j + u) * 16 + ln;
#pragma unroll
            for (int r = 0; r < 8; ++r) {
                float v = D[r];
                if (v > best[r]) { best[r] = v; bidx[r] = depth; }
            }
        }
    }
}

// ---------------------------------------------------------------------------
// Kernel D: fused scores-GEMM + argmax with V_WMMA_F32_16X16X4_F32.
// One wave -> 16 pixels. A = norm_b tile (16x4), B = codebook slice (4x16).
// Codebook chunks double-buffered in LDS via GLOBAL_LOAD_ASYNC_TO_LDS_B128
// (ASYNCcnt) so chunk c+1 streams in while WMMAs run on chunk c.
// ---------------------------------------------------------------------------
__global__ __launch_bounds__(256) void argmax_kernel(const float* __restrict__ normW,
                                                     const float* __restrict__ normB,
                                                     float* __restrict__ out) {
    __shared__ float4 sCB[2][CHUNK];               // 62.5 KB double buffer
    const int tid      = threadIdx.x;
    const int lane     = tid & 31;
    const int ln       = lane & 15;
    const bool hi      = lane >= 16;
    const int koff     = hi ? 2 : 0;
    const int wave     = blockIdx.x * 8 + (tid >> 5);
    const int pixBase  = wave * 16;

    // A operand: per-lane float2 of this wave's pixel tile
    const float* ap = normB + (pixBase + ln) * 4 + koff;
    v2f A; A.x = ap[0]; A.y = ap[1];

    float best[8]; int bidx[8];
#pragma unroll
    for (int r = 0; r < 8; ++r) { best[r] = -3.0e38f; bidx[r] = 0; }

    const float4* cb = (const float4*)normW;
    const int NCH = NN / CHUNK;                    // 5

#if HAVE_ASYNC_LDS
    stage_chunk(cb, &sCB[0][0], tid);              // prologue: chunk 0
    wait_async0();
    __syncthreads();
    for (int c = 0; c < NCH; ++c) {
        if (c + 1 < NCH) stage_chunk(cb + (c + 1) * CHUNK, &sCB[(c + 1) & 1][0], tid);
        wmma_chunk((const float*)&sCB[c & 1][0], A, ln, koff, c * CHUNK, best, bidx);
        wait_async0();                             // chunk c+1 landed in LDS
        __syncthreads();
    }
#else
    for (int c = 0; c < NCH; ++c) {
        stage_chunk(cb + c * CHUNK, &sCB[0][0], tid);
        __syncthreads();
        wmma_chunk((const float*)&sCB[0][0], A, ln, koff, c * CHUNK, best, bidx);
        __syncthreads();
    }
#endif

    // cross-lane arg-reduction within each 16-lane half (first-max semantics)
#pragma unroll
    for (int r = 0; r < 8; ++r) {
        float v = best[r]; int ix = bidx[r];
        for (int m = 1; m < 16; m <<= 1) {
            float ov = __shfl_xor(v, m, 32);
            int   oi = __shfl_xor(ix, m, 32);
            bool take = (ov > v) | ((ov == v) & (oi < ix));   // branch-free
            v  = take ? ov : v;
            ix = take ? oi : ix;
        }
        if (ln == 0) out[pixBase + r + (hi ? 8 : 0)] = (float)ix;
    }
}

// ---------------------------------------------------------------------------
extern "C" void kernel_launch(void* const* d_in, const int* in_sizes, int n_in,
                              void* d_out, int out_size, void* d_ws, size_t ws_size,
                              hipStream_t stream) {
    (void)in_sizes; (void)n_in; (void)out_size; (void)ws_size;
    const float* ModFs   = (const float*)d_in[0];
    const float* DemodFs = (const float*)d_in[1];
    const int*   gt      = (const int*)d_in[2];
    const float* noise   = (const float*)d_in[3];
    float*       out     = (float*)d_out;

    float* ws    = (float*)d_ws;
    float* stats = ws;                      // 16 floats (use 4)
    float* corrW = ws + 16;                 // NN*4 floats
    float* normW = corrW + NN * 4;          // NN*4 floats
    float* normB = normW + NN * 4;          // NPIX*4 floats

    stats_kernel<<<1, 256, 0, stream>>>(ModFs, DemodFs, stats);
    corr_kernel<<<(NN + 255) / 256, 256, 0, stream>>>(ModFs, DemodFs, stats, corrW, normW);
    pix_kernel<<<(NPIX + 255) / 256, 256, 0, stream>>>(gt, noise, stats, corrW, normB);
    argmax_kernel<<<NPIX / 128, 256, 0, stream>>>(normW, normB, out);
}